// SAGE_48086453846609
// MI455X (gfx1250) — compile-verified
//
#include <hip/hip_runtime.h>
#include <math.h>

typedef float v2f __attribute__((ext_vector_type(2)));
typedef float v8f __attribute__((ext_vector_type(8)));

#define K_DIM 128

// ---------------------------------------------------------------- utilities
__global__ void zero_f32(float* __restrict__ p, size_t n) {
    size_t i = (size_t)blockIdx.x * blockDim.x + threadIdx.x;
    size_t stride = (size_t)gridDim.x * blockDim.x;
    for (; i < n; i += stride) p[i] = 0.0f;
}

// one thread per edge: deg[dst] += 1  (degree identical for all 3 layers)
__global__ void deg_count(const int* __restrict__ dst, float* __restrict__ deg, int E) {
    int e = blockIdx.x * blockDim.x + threadIdx.x;
    if (e < E) atomicAdd(&deg[dst[e]], 1.0f);
}

// one wave (32 lanes) per edge, 4 channels per lane (128 ch total).
// float4 gather is coalesced (512B per wave); atomic f32 adds resolve at L2.
__global__ void scatter_add(const float* __restrict__ h, const int* __restrict__ src,
                            const int* __restrict__ dst, float* __restrict__ agg, int E) {
    int gid  = blockIdx.x * blockDim.x + threadIdx.x;
    int e    = gid >> 5;
    int lane = gid & 31;
    if (e >= E) return;
    int s = src[e];
    int d = dst[e];
    const float4 v = *(const float4*)(h + (size_t)s * K_DIM + lane * 4);
    float* out = agg + (size_t)d * K_DIM + lane * 4;
    atomicAdd(out + 0, v.x);
    atomicAdd(out + 1, v.y);
    atomicAdd(out + 2, v.z);
    atomicAdd(out + 3, v.w);
}

__device__ __forceinline__ v8f wmma_f32(v2f a, v2f b, v8f c) {
    // V_WMMA_F32_16X16X4_F32 : D = A(16x4) x B(4x16) + C(16x16), full fp32
    return __builtin_amdgcn_wmma_f32_16x16x4_f32(
        /*neg_a=*/false, a, /*neg_b=*/false, b,
        /*c_mod=*/(short)0, c, /*reuse_a=*/false, /*reuse_b=*/false);
}

// out[m][n] = sum_k mean[m][k]*Wl[n][k] + sum_k xin[m][k]*Wr[n][k] + bias[n]
// mean[m][k] = agg[m][k] / max(deg[m],1).  One wave per 16-row M-tile,
// NT n-tiles of 16 accumulated in registers (NT*8 VGPRs of C).
template <int NT, int NOUT, bool RELU>
__global__ __launch_bounds__(128) void sage_layer(
        const float* __restrict__ agg, const float* __restrict__ deg,
        const float* __restrict__ xin,
        const float* __restrict__ Wl, const float* __restrict__ Wr,
        const float* __restrict__ bias, float* __restrict__ out, int M) {
    int wave = blockIdx.x * (blockDim.x >> 5) + (threadIdx.x >> 5);
    int lane = threadIdx.x & 31;
    int m0 = wave * 16;
    if (m0 >= M) return;                 // wave-uniform: EXEC all-1s for WMMA

    int half = lane >> 4;                // 0: K 0..1 | 1: K 2..3 (A & B frags)
    int lr   = lane & 15;                // row-in-tile (A) / col-in-tile (B)
    int koff = half * 2;
    int row  = m0 + lr;

    float invd = 1.0f / fmaxf(deg[row], 1.0f);
    const float* arow = agg + (size_t)row * K_DIM;
    const float* xrow = xin + (size_t)row * K_DIM;

    v8f acc[NT] = {};

    // pass 1: mean @ Wl.T
#pragma unroll 4
    for (int k = 0; k < K_DIM; k += 4) {
        float2 av = *(const float2*)(arow + k + koff);
        v2f a; a.x = av.x * invd; a.y = av.y * invd;
#pragma unroll
        for (int t = 0; t < NT; ++t) {
            int col = t * 16 + lr;
            v2f b;
            if (NOUT == 128 || col < NOUT) {
                float2 w = *(const float2*)(Wl + (size_t)col * K_DIM + k + koff);
                b.x = w.x; b.y = w.y;
            } else { b.x = 0.0f; b.y = 0.0f; }
            acc[t] = wmma_f32(a, b, acc[t]);
        }
    }
    // pass 2: xin @ Wr.T
#pragma unroll 4
    for (int k = 0; k < K_DIM; k += 4) {
        float2 av = *(const float2*)(xrow + k + koff);
        v2f a; a.x = av.x; a.y = av.y;
#pragma unroll
        for (int t = 0; t < NT; ++t) {
            int col = t * 16 + lr;
            v2f b;
            if (NOUT == 128 || col < NOUT) {
                float2 w = *(const float2*)(Wr + (size_t)col * K_DIM + k + koff);
                b.x = w.x; b.y = w.y;
            } else { b.x = 0.0f; b.y = 0.0f; }
            acc[t] = wmma_f32(a, b, acc[t]);
        }
    }

    // epilogue: C vgpr j -> row m0 + j + 8*half, col = t*16 + lr
    int rbase = m0 + half * 8;
#pragma unroll
    for (int t = 0; t < NT; ++t) {
        int col = t * 16 + lr;
        bool cok = (NOUT == 128) || (col < NOUT);
        float bv = cok ? bias[col] : 0.0f;
#pragma unroll
        for (int j = 0; j < 8; ++j) {
            float v = acc[t][j] + bv;
            if (RELU) v = fmaxf(v, 0.0f);
            if (cok) out[(size_t)(rbase + j) * NOUT + col] = v;
        }
    }
}

// in-place log-softmax per row (C=40, tiny: 16MB total traffic)
__global__ void log_softmax_rows(float* __restrict__ out, int M, int C) {
    int r = blockIdx.x * blockDim.x + threadIdx.x;
    if (r >= M) return;
    float* p = out + (size_t)r * C;
    float mx = -3.402823466e38f;
    for (int c = 0; c < C; ++c) mx = fmaxf(mx, p[c]);
    float s = 0.0f;
    for (int c = 0; c < C; ++c) s += expf(p[c] - mx);
    float lse = logf(s) + mx;
    for (int c = 0; c < C; ++c) p[c] = p[c] - lse;
}

// ---------------------------------------------------------------- driver
extern "C" void kernel_launch(void* const* d_in, const int* in_sizes, int n_in,
                              void* d_out, int out_size, void* d_ws, size_t ws_size,
                              hipStream_t stream) {
    const float* x   = (const float*)d_in[0];
    const int*   ei  = (const int*)  d_in[1];
    const float* Wl1 = (const float*)d_in[2];
    const float* Wr1 = (const float*)d_in[3];
    const float* b1  = (const float*)d_in[4];
    const float* Wl2 = (const float*)d_in[5];
    const float* Wr2 = (const float*)d_in[6];
    const float* b2  = (const float*)d_in[7];
    const float* Wl3 = (const float*)d_in[8];
    const float* Wr3 = (const float*)d_in[9];
    const float* b3  = (const float*)d_in[10];

    const int N = in_sizes[0] / K_DIM;     // 100000
    const int E = in_sizes[1] / 2;         // 1600000
    const int OUTC = 40;
    const int* src = ei;
    const int* dst = ei + E;

    // workspace layout (256B aligned)
    char* ws = (char*)d_ws;
    size_t degBytes = ((size_t)N * sizeof(float) + 255) & ~(size_t)255;
    float* deg = (float*)ws;
    float* agg = (float*)(ws + degBytes);
    float* h   = (float*)(ws + degBytes + (size_t)N * K_DIM * sizeof(float));

    const size_t aggN = (size_t)N * K_DIM;
    const int ZB = 2048, TB = 256;
    const int edgeBlocks  = (E + TB - 1) / TB;
    const int scatBlocks  = ((E * 32) + TB - 1) / TB;
    const int tiles       = (N + 15) / 16;            // 6250
    const int gemmBlocks  = (tiles + 3) / 4;          // 4 waves / block
    const int lsmBlocks   = (N + TB - 1) / TB;

    // degrees (once)
    zero_f32<<<ZB, TB, 0, stream>>>(deg, (size_t)N);
    deg_count<<<edgeBlocks, TB, 0, stream>>>(dst, deg, E);

    // layer 1: agg = scatter(x); h = relu(mean@Wl1.T + x@Wr1.T + b1)
    zero_f32<<<ZB, TB, 0, stream>>>(agg, aggN);
    scatter_add<<<scatBlocks, TB, 0, stream>>>(x, src, dst, agg, E);
    sage_layer<8, 128, true><<<gemmBlocks, 128, 0, stream>>>(agg, deg, x, Wl1, Wr1, b1, h, N);

    // layer 2 (in-place on h is safe: agg fully materialized first)
    zero_f32<<<ZB, TB, 0, stream>>>(agg, aggN);
    scatter_add<<<scatBlocks, TB, 0, stream>>>(h, src, dst, agg, E);
    sage_layer<8, 128, true><<<gemmBlocks, 128, 0, stream>>>(agg, deg, h, Wl2, Wr2, b2, h, N);

    // layer 3 -> logits into d_out, then in-place log-softmax
    zero_f32<<<ZB, TB, 0, stream>>>(agg, aggN);
    scatter_add<<<scatBlocks, TB, 0, stream>>>(h, src, dst, agg, E);
    sage_layer<3, 40, false><<<gemmBlocks, 128, 0, stream>>>(agg, deg, h, Wl3, Wr3, b3,
                                                             (float*)d_out, N);
    log_softmax_rows<<<lsmBlocks, TB, 0, stream>>>((float*)d_out, N, OUTC);
}